// SpikingLayer_27736898798060
// MI455X (gfx1250) — compile-verified
//
#include <hip/hip_runtime.h>
#include <cstdint>
#include <cstddef>

// Problem constants (from reference): x is (B, C, T) float32, scan over T.
namespace {
constexpr int kB    = 256;
constexpr int kC    = 512;
constexpr int kT    = 512;
constexpr int kRows = kB * kC;          // 131072 independent scan rows
constexpr int RPB   = 256;              // rows per block == blockDim.x
constexpr int TC    = 16;               // time-chunk per row (floats) = 64 B
constexpr int NCH   = kT / TC;          // 32 chunks
constexpr int STRIDE = TC + 4;          // 20 floats = 80 B row pitch (16B aligned)
}

__global__ __launch_bounds__(RPB) void snn_scan_kernel(
    const float* __restrict__ x,
    const float* __restrict__ beta_logit,
    const float* __restrict__ threshold,
    const float* __restrict__ init_rand,
    float* __restrict__ out)
{
  // Double-buffered input tile + output staging tile (60 KB total).
  __shared__ __align__(16) float xs[2][RPB * STRIDE];
  __shared__ __align__(16) float ss[RPB * STRIDE];

  const int tid   = threadIdx.x;
  const int row0  = blockIdx.x * RPB;
  const int myrow = row0 + tid;
  const int ch    = myrow & (kC - 1);   // channel index (C = 512, rows are b*C + c)

  // Per-row parameters.
  const float th   = threshold[ch];
  const float beta = 1.0f / (1.0f + expf(-beta_logit[ch]));
  float mem        = init_rand[myrow] * th;

  // Staging lane mapping: 4 lanes per row (TC/4 float4 chunks), 64 rows / step.
  const int lrow = tid >> 2;            // 0..63
  const int lcol = tid & 3;             // which float4 within the 16-float chunk

  const uint32_t lds_x0 = (uint32_t)(uintptr_t)&xs[0][0];
  const uint32_t lds_x1 = (uint32_t)(uintptr_t)&xs[1][0];

  // Issue one chunk's worth of async global->LDS copies (4 instrs per wave,
  // 512 B each, tracked by ASYNCcnt; loads complete in order).
  auto issue_async_loads = [&](int ci, uint32_t lds_base) {
#pragma unroll
    for (int it = 0; it < 4; ++it) {
      const int r = it * 64 + lrow;
      const uint32_t lds_addr = lds_base + (uint32_t)((r * STRIDE + lcol * 4) * 4);
      const uint32_t goff     = (uint32_t)(((row0 + r) * kT + ci * TC + lcol * 4) * 4);
      asm volatile("global_load_async_to_lds_b128 %0, %1, %2"
                   :: "v"(lds_addr), "v"(goff), "s"(x)
                   : "memory");
    }
  };

  // Prologue: prefetch chunk 0.
  issue_async_loads(0, lds_x0);

  for (int ci = 0; ci < NCH; ++ci) {
    const float* curp = (ci & 1) ? &xs[1][0] : &xs[0][0];

    if (ci + 1 < NCH) {
      // Prefetch next chunk into the other buffer, then wait for the current
      // chunk only: async loads retire in order, so ASYNCcnt <= 4 implies the
      // older 4 (current chunk) have landed in LDS.
      issue_async_loads(ci + 1, (ci & 1) ? lds_x0 : lds_x1);
      asm volatile("s_wait_asynccnt 0x4" ::: "memory");
    } else {
      asm volatile("s_wait_asynccnt 0x0" ::: "memory");
    }
    __syncthreads();   // all waves' async loads for this chunk are visible

    // Serial LIF scan over this chunk: 4 x float4 from LDS, spikes to LDS.
    const float4* xr = (const float4*)(curp + tid * STRIDE);
    float4*       sr = (float4*)(&ss[tid * STRIDE]);
#pragma unroll
    for (int q = 0; q < TC / 4; ++q) {
      const float4 xv = xr[q];
      float4 sv;
      { const float m = fmaf(beta, mem, xv.x); const bool s = (m >= th); sv.x = s ? 1.f : 0.f; mem = s ? (m - th) : m; }
      { const float m = fmaf(beta, mem, xv.y); const bool s = (m >= th); sv.y = s ? 1.f : 0.f; mem = s ? (m - th) : m; }
      { const float m = fmaf(beta, mem, xv.z); const bool s = (m >= th); sv.z = s ? 1.f : 0.f; mem = s ? (m - th) : m; }
      { const float m = fmaf(beta, mem, xv.w); const bool s = (m >= th); sv.w = s ? 1.f : 0.f; mem = s ? (m - th) : m; }
      sr[q] = sv;
    }
    __syncthreads();   // ss tile complete before transposed store-out

    // Coalesced store-out of the spike tile (64 B contiguous per 4 lanes).
#pragma unroll
    for (int it = 0; it < 4; ++it) {
      const int r = it * 64 + lrow;
      const float4 v = *(const float4*)(&ss[r * STRIDE + lcol * 4]);
      *(float4*)(&out[(size_t)(row0 + r) * kT + ci * TC + lcol * 4]) = v;
    }
    // Next iteration's top barrier (after its own asynccnt wait) separates
    // these LDS reads from the next chunk's ss writes, and the bottom barrier
    // above separates xs reads from the re-use of this chunk's buffer.
  }
}

extern "C" void kernel_launch(void* const* d_in, const int* in_sizes, int n_in,
                              void* d_out, int out_size, void* d_ws, size_t ws_size,
                              hipStream_t stream) {
  (void)in_sizes; (void)n_in; (void)out_size; (void)d_ws; (void)ws_size;
  const float* x  = (const float*)d_in[0];  // (B, C, T)
  const float* bl = (const float*)d_in[1];  // (C,) beta logits
  const float* th = (const float*)d_in[2];  // (C,) thresholds
  const float* ir = (const float*)d_in[3];  // (B, C) init_rand
  float* out = (float*)d_out;               // (B, C, T) spikes

  dim3 grid(kRows / RPB);   // 512 blocks
  dim3 block(RPB);          // 256 threads = 8 wave32s
  hipLaunchKernelGGL(snn_scan_kernel, grid, block, 0, stream, x, bl, th, ir, out);
}